// HypergraphEmissionModel_56547539419452
// MI455X (gfx1250) — compile-verified
//
#include <hip/hip_runtime.h>
#include <hip/hip_bf16.h>

#define N_NODES 100000
#define N_EDGES 200000
#define NNZ_CNT 1200000
#define DIM 64

typedef float v2f __attribute__((ext_vector_type(2)));
typedef float v8f __attribute__((ext_vector_type(8)));

// ---------------------------------------------------------------------------
// Utility fills
// ---------------------------------------------------------------------------
__global__ __launch_bounds__(256) void fill_f32(float* __restrict__ p, float v, int n) {
    int i = blockIdx.x * blockDim.x + threadIdx.x;
    if (i < n) p[i] = v;
}

// out[r, c] = bias[c]   (row-major N x 64); bias is the accumulator seed.
__global__ __launch_bounds__(256) void init_bias(float* __restrict__ out,
                                                 const float* __restrict__ b, int n) {
    int i = blockIdx.x * blockDim.x + threadIdx.x;
    if (i < n) out[i] = b[i & (DIM - 1)];
}

// ---------------------------------------------------------------------------
// Degree / hyperedge-size histograms
// ---------------------------------------------------------------------------
__global__ __launch_bounds__(256) void degrees(const int* __restrict__ nidx,
                                               const int* __restrict__ eidx,
                                               float* __restrict__ D,
                                               float* __restrict__ B, int nnz) {
    int i = blockIdx.x * blockDim.x + threadIdx.x;
    if (i < nnz) {
        atomicAdd(&D[nidx[i]], 1.0f);
        atomicAdd(&B[eidx[i]], 1.0f);
    }
}

// ---------------------------------------------------------------------------
// dst[sidx[i], :] += (1/counts[sidx[i]]) * src[gidx[i], :]
// 16 lanes per nonzero, 4 cols (one float4) per lane -> fully coalesced 256B row.
// Scaling by the destination's inverse count folds B^-1 / D^-1 into the pass.
// ---------------------------------------------------------------------------
__global__ __launch_bounds__(256) void scatter_scaled(float* __restrict__ dst,
                                                      const float* __restrict__ src,
                                                      const int* __restrict__ gidx,
                                                      const int* __restrict__ sidx,
                                                      const float* __restrict__ counts,
                                                      int nnz) {
    long long T = (long long)blockIdx.x * blockDim.x + threadIdx.x;
    int i = (int)(T >> 4);
    if (i >= nnz) return;
    int c  = ((int)T & 15) * 4;
    int g  = gidx[i];
    int s  = sidx[i];
    float cnt = counts[s];
    float inv = cnt > 0.0f ? 1.0f / cnt : 0.0f;
    const float4 v = *(const float4*)(src + (long long)g * DIM + c);
    float* d = dst + (long long)s * DIM + c;
    atomicAdd(d + 0, inv * v.x);
    atomicAdd(d + 1, inv * v.y);
    atomicAdd(d + 2, inv * v.z);
    atomicAdd(d + 3, inv * v.w);
}

// ---------------------------------------------------------------------------
// Y[nrows x 64] = (relu_in ? relu(X) : X) @ W[64 x 64], fp32 WMMA 16x16x4.
// One wave per 16-row tile; 4 accumulators (v8f) cover the 64 output columns.
// A layout (16x4 f32): VGPR j <- K = j + 2*(lane>=16), row = lane&15 (ISA 7.12.2)
// B layout (4x16 f32): VGPR j <- K = j + 2*(lane>=16), col = lane&15
// C/D layout:          VGPR r <- M = r + 8*(lane>=16), N = lane&15
// ---------------------------------------------------------------------------
__global__ __launch_bounds__(256) void gemm64_wmma_f32(const float* __restrict__ X,
                                                       const float* __restrict__ W,
                                                       float* __restrict__ Y,
                                                       int nrows, int relu_in) {
    const int lane = threadIdx.x & 31;
    const int wave = threadIdx.x >> 5;
    const int m0   = (blockIdx.x * 8 + wave) * 16;
    if (m0 >= nrows) return;                       // wave-uniform: EXEC stays all-1s

    const int half = lane >> 4;                    // 0 or 1
    const int l16  = lane & 15;
    const long long arow = (long long)(m0 + l16) * DIM;

    v8f acc[4];
#pragma unroll
    for (int t = 0; t < 4; ++t) acc[t] = (v8f){0.f, 0.f, 0.f, 0.f, 0.f, 0.f, 0.f, 0.f};

#pragma unroll
    for (int k0 = 0; k0 < DIM; k0 += 4) {
        const int ka = k0 + 2 * half;
        float a0 = X[arow + ka + 0];
        float a1 = X[arow + ka + 1];
        if (relu_in) {
            a0 = a0 > 0.f ? a0 : 0.f;
            a1 = a1 > 0.f ? a1 : 0.f;
        }
        v2f a;
        a.x = a0;
        a.y = a1;
#pragma unroll
        for (int t = 0; t < 4; ++t) {
            const int n0 = t * 16;
            v2f b;
            b.x = W[(ka + 0) * DIM + n0 + l16];
            b.y = W[(ka + 1) * DIM + n0 + l16];
            acc[t] = __builtin_amdgcn_wmma_f32_16x16x4_f32(
                false, a, false, b, (short)0, acc[t], false, false);
        }
    }

#pragma unroll
    for (int t = 0; t < 4; ++t) {
        const int n0 = t * 16;
#pragma unroll
        for (int r = 0; r < 8; ++r) {
            Y[(long long)(m0 + r + 8 * half) * DIM + n0 + l16] = acc[t][r];
        }
    }
}

// ---------------------------------------------------------------------------
// Host-side orchestration
// ---------------------------------------------------------------------------
extern "C" void kernel_launch(void* const* d_in, const int* in_sizes, int n_in,
                              void* d_out, int out_size, void* d_ws, size_t ws_size,
                              hipStream_t stream) {
    const float* x   = (const float*)d_in[0];
    const int*   hei = (const int*)d_in[1];     // [2, NNZ] stacked
    const float* W1  = (const float*)d_in[2];
    const float* b1  = (const float*)d_in[3];
    const float* W2  = (const float*)d_in[4];
    const float* b2  = (const float*)d_in[5];

    const int* node_idx = hei;                  // row 0
    const int* edge_idx = hei + NNZ_CNT;        // row 1

    float* ws   = (float*)d_ws;
    float* D    = ws;                                   // N_NODES
    float* B    = D + N_NODES;                          // N_EDGES
    float* xt   = B + N_EDGES;                          // N_NODES*64
    float* ebuf = xt + (size_t)N_NODES * DIM;           // N_EDGES*64
    float* hbuf = ebuf + (size_t)N_EDGES * DIM;         // N_NODES*64
    float* y    = (float*)d_out;                        // N_NODES*64

    const int nodeElems = N_NODES * DIM;                // 6.4M
    const int edgeElems = N_EDGES * DIM;                // 12.8M

    auto blocks = [](long long n, int bs) { return (int)((n + bs - 1) / bs); };
    const int gemmBlocks    = blocks((N_NODES / 16 + 7) / 8 * 8, 8); // 782
    const int scatterBlocks = blocks((long long)NNZ_CNT * 16, 256);  // 75000

    // Degrees
    fill_f32<<<blocks(N_NODES, 256), 256, 0, stream>>>(D, 0.f, N_NODES);
    fill_f32<<<blocks(N_EDGES, 256), 256, 0, stream>>>(B, 0.f, N_EDGES);
    degrees<<<blocks(NNZ_CNT, 256), 256, 0, stream>>>(node_idx, edge_idx, D, B, NNZ_CNT);

    // ----- Layer 1 -----
    gemm64_wmma_f32<<<gemmBlocks, 256, 0, stream>>>(x, W1, xt, N_NODES, 0);
    fill_f32<<<blocks(edgeElems, 256), 256, 0, stream>>>(ebuf, 0.f, edgeElems);
    // e = B^-1 * segsum_edge(xt[node])
    scatter_scaled<<<scatterBlocks, 256, 0, stream>>>(ebuf, xt, node_idx, edge_idx, B, NNZ_CNT);
    // h = b1 + D^-1 * segsum_node(e[edge])
    init_bias<<<blocks(nodeElems, 256), 256, 0, stream>>>(hbuf, b1, nodeElems);
    scatter_scaled<<<scatterBlocks, 256, 0, stream>>>(hbuf, ebuf, edge_idx, node_idx, D, NNZ_CNT);

    // ----- Layer 2 (ReLU folded into the GEMM's A-load) -----
    gemm64_wmma_f32<<<gemmBlocks, 256, 0, stream>>>(hbuf, W2, xt, N_NODES, 1);
    fill_f32<<<blocks(edgeElems, 256), 256, 0, stream>>>(ebuf, 0.f, edgeElems);
    scatter_scaled<<<scatterBlocks, 256, 0, stream>>>(ebuf, xt, node_idx, edge_idx, B, NNZ_CNT);
    init_bias<<<blocks(nodeElems, 256), 256, 0, stream>>>(y, b2, nodeElems);
    scatter_scaled<<<scatterBlocks, 256, 0, stream>>>(y, ebuf, edge_idx, node_idx, D, NNZ_CNT);
}